// RNNPool_82334523064946
// MI455X (gfx1250) — compile-verified
//
#include <hip/hip_runtime.h>
#include <hip/hip_bf16.h>

typedef float v2f __attribute__((ext_vector_type(2)));
typedef float v8f __attribute__((ext_vector_type(8)));

#define OH 27
#define OW 27
#define HWDIM 112
#define CIN 16

__device__ __forceinline__ float fast_rcp(float x) {
    return __builtin_amdgcn_rcpf(x);          // v_rcp_f32
}
__device__ __forceinline__ float sigmoidf_(float x) {
    // branchless; x=+inf -> 1, x=-inf -> 0
    return fast_rcp(1.0f + __expf(-x));
}
__device__ __forceinline__ float tanhf_(float x) {
    // tanh(x) = 1 - 2/(e^{2x}+1); branchless, correct limits at +/-inf
    return 1.0f - 2.0f * fast_rcp(1.0f + __expf(2.0f * x));
}

__device__ __forceinline__ v8f wmma4_f32(v2f a, v2f b, v8f c) {
    // D = A(16x4,f32) * B(4x16,f32) + C(16x16,f32)
    return __builtin_amdgcn_wmma_f32_16x16x4_f32(
        false, a, false, b, (short)0, c, false, false);
}

__global__ __launch_bounds__(32) void rnnpool_kernel(
    const float* __restrict__ inp,
    const float* __restrict__ W1, const float* __restrict__ U1,
    const float* __restrict__ bg1, const float* __restrict__ bu1,
    const float* __restrict__ zeta1, const float* __restrict__ nu1,
    const float* __restrict__ W2, const float* __restrict__ U2,
    const float* __restrict__ bg2, const float* __restrict__ bu2,
    const float* __restrict__ zeta2, const float* __restrict__ nu2,
    float* __restrict__ out, int numPatches)
{
    __shared__ float lds1[256];   // 16x16 h exchange / final stage-A hiddens
    __shared__ float lds2[256];   // 16x16 h2 exchange for stage B

    const int lane = threadIdx.x;     // 0..31 (wave32)
    const int n    = lane & 15;       // N column (and M row for A-reads)
    const int half = lane >> 4;       // 0 or 1

    const int p = blockIdx.x;
    if (p >= numPatches) return;
    const int bimg = p / (OH * OW);
    const int rem  = p % (OH * OW);
    const int i0 = (rem / OW) * 4;    // top row of patch
    const int j0 = (rem % OW) * 4;    // left col of patch

    // ---- preload weights into WMMA B-layout registers ----
    // B[j][lane] = W[k][n], k = 4*b + j + 2*half, n = lane%16
    v2f bW1[4], bU1[2], bW2[2], bU2[4];
#pragma unroll
    for (int b = 0; b < 4; b++) {
#pragma unroll
        for (int j = 0; j < 2; j++) {
            int k = 4 * b + j + 2 * half;          // 0..15
            bW1[b][j] = (n < 8) ? W1[k * 8 + n] : 0.0f;   // W1: (16,8), N-padded
            bU2[b][j] = U2[k * 16 + n];                   // U2: (16,16)
        }
    }
#pragma unroll
    for (int b = 0; b < 2; b++) {
#pragma unroll
        for (int j = 0; j < 2; j++) {
            int k = 4 * b + j + 2 * half;          // 0..7
            bU1[b][j] = (n < 8) ? U1[k * 8 + n] : 0.0f;   // U1: (8,8), N-padded
            bW2[b][j] = W2[k * 16 + n];                   // W2: (8,16)
        }
    }

    const float bg1v = (n < 8) ? bg1[n] : 0.0f;   // padded cols: z=0.5,c=0 -> h stays 0
    const float bu1v = (n < 8) ? bu1[n] : 0.0f;
    const float bg2v = bg2[n];
    const float bu2v = bu2[n];
    const float zg1 = sigmoidf_(zeta1[0]);
    const float ng1 = sigmoidf_(nu1[0]);
    const float zg2 = sigmoidf_(zeta2[0]);
    const float ng2 = sigmoidf_(nu2[0]);

    // per-lane stage-A pixel walk: row-seqs (m<8) step along x; col-seqs step along y
    const int m = n;
    const int y0 = (m < 8) ? (i0 + m) : i0;
    const int x0 = (m < 8) ? j0 : (j0 + m - 8);
    const float* aptr = inp + (((size_t)bimg * HWDIM + y0) * HWDIM + x0) * CIN + 2 * half;
    const int pixstride = (m < 8) ? CIN : (HWDIM * CIN);
    const int ldsA = n * 16 + 2 * half;           // A-read base into 16x16 tiles
    const int ldsD = (8 * half) * 16 + n;         // D-write base

    // ================= Stage A: 16 GRNN1 sequences (8 rows + 8 cols) ==========
    // D/C layout: element (M = v + 8*half, N = n) in h[v]
    v8f h = {};
    // preloaded X_t A-fragments (distinct regs -> clause-batched loads)
    v2f ax0 = *(const v2f*)(aptr + 0);
    v2f ax1 = *(const v2f*)(aptr + 4);
    v2f ax2 = *(const v2f*)(aptr + 8);
    v2f ax3 = *(const v2f*)(aptr + 12);

#pragma unroll 1
    for (int t = 0; t < 8; t++) {
        // publish h (D layout) as row-major 16x16 tile
#pragma unroll
        for (int v = 0; v < 8; v++) lds1[ldsD + v * 16] = h[v];
        __syncthreads();

        // h @ U1 (K=8) on an independent accumulator chain
        const v2f ah0 = *(const v2f*)(&lds1[ldsA + 0]);
        const v2f ah1 = *(const v2f*)(&lds1[ldsA + 4]);

        v8f accx = {};
        accx = wmma4_f32(ax0, bW1[0], accx);
        accx = wmma4_f32(ax1, bW1[1], accx);
        accx = wmma4_f32(ax2, bW1[2], accx);
        accx = wmma4_f32(ax3, bW1[3], accx);

        v8f accu = {};
        accu = wmma4_f32(ah0, bU1[0], accu);
        accu = wmma4_f32(ah1, bU1[1], accu);

        // prefetch X_{t+1} while WMMAs drain (uniform branch, no OOB)
        if (t < 7) {
            aptr += pixstride;
            ax0 = *(const v2f*)(aptr + 0);
            ax1 = *(const v2f*)(aptr + 4);
            ax2 = *(const v2f*)(aptr + 8);
            ax3 = *(const v2f*)(aptr + 12);
        }
        __syncthreads();

        // elementwise gate update (branchless transcendentals)
#pragma unroll
        for (int v = 0; v < 8; v++) {
            float pre = accx[v] + accu[v];
            float z = sigmoidf_(pre + bg1v);
            float c = tanhf_(pre + bu1v);
            h[v] = z * h[v] + (zg1 * (1.0f - z) + ng1) * c;
        }
    }
    // publish final stage-A hiddens: rows 0-7 = h_row[r], rows 8-15 = h_col[c]
#pragma unroll
    for (int v = 0; v < 8; v++) lds1[ldsD + v * 16] = h[v];
    __syncthreads();

    // ================= Stage B: 4 GRNN2 sequences (o21,o22,o23,o24) ===========
    v8f h2 = {};
#pragma unroll 1
    for (int t = 0; t < 8; t++) {
#pragma unroll
        for (int v = 0; v < 8; v++) lds2[ldsD + v * 16] = h2[v];
        __syncthreads();

        // x2 @ W2, K = 8; row m input: m=0 h[t], m=1 h[7-t], m=2 hc[t], m=3 hc[7-t]
        const int srcrow = (m < 4)
            ? ((m >> 1) * 8 + ((m & 1) ? (7 - t) : t))
            : 0;
        const int srcbase = srcrow * 16 + 2 * half;
        v2f aw0, aw1;
        if (m < 4) {
            aw0 = *(const v2f*)(&lds1[srcbase + 0]);
            aw1 = *(const v2f*)(&lds1[srcbase + 4]);
        } else {
            aw0[0] = 0.0f; aw0[1] = 0.0f;
            aw1[0] = 0.0f; aw1[1] = 0.0f;
        }
        const v2f ah0 = *(const v2f*)(&lds2[ldsA + 0]);
        const v2f ah1 = *(const v2f*)(&lds2[ldsA + 4]);
        const v2f ah2 = *(const v2f*)(&lds2[ldsA + 8]);
        const v2f ah3 = *(const v2f*)(&lds2[ldsA + 12]);

        v8f accw = {};
        accw = wmma4_f32(aw0, bW2[0], accw);
        accw = wmma4_f32(aw1, bW2[1], accw);

        v8f accu = {};
        accu = wmma4_f32(ah0, bU2[0], accu);
        accu = wmma4_f32(ah1, bU2[1], accu);
        accu = wmma4_f32(ah2, bU2[2], accu);
        accu = wmma4_f32(ah3, bU2[3], accu);
        __syncthreads();

#pragma unroll
        for (int v = 0; v < 8; v++) {
            float pre = accw[v] + accu[v];
            float z = sigmoidf_(pre + bg2v);
            float c = tanhf_(pre + bu2v);
            h2[v] = z * h2[v] + (zg2 * (1.0f - z) + ng2) * c;
        }
    }

    // write out[p*64 + m*16 + n] for m = 0..3 (lanes 0-15 hold m = v)
    if (half == 0) {
        float* op = out + (size_t)p * 64 + n;
        op[0]  = h2[0];
        op[16] = h2[1];
        op[32] = h2[2];
        op[48] = h2[3];
    }
}

extern "C" void kernel_launch(void* const* d_in, const int* in_sizes, int n_in,
                              void* d_out, int out_size, void* d_ws, size_t ws_size,
                              hipStream_t stream) {
    const float* inp   = (const float*)d_in[0];
    const float* W1    = (const float*)d_in[1];
    const float* U1    = (const float*)d_in[2];
    const float* bg1   = (const float*)d_in[3];
    const float* bu1   = (const float*)d_in[4];
    const float* zeta1 = (const float*)d_in[5];
    const float* nu1   = (const float*)d_in[6];
    const float* W2    = (const float*)d_in[7];
    const float* U2    = (const float*)d_in[8];
    const float* bg2   = (const float*)d_in[9];
    const float* bu2   = (const float*)d_in[10];
    const float* zeta2 = (const float*)d_in[11];
    const float* nu2   = (const float*)d_in[12];
    float* out = (float*)d_out;

    const int numPatches = out_size / 64;   // B*27*27
    rnnpool_kernel<<<numPatches, 32, 0, stream>>>(
        inp, W1, U1, bg1, bu1, zeta1, nu1,
        W2, U2, bg2, bu2, zeta2, nu2, out, numPatches);
}